// CustomLossTarget_33775622816507
// MI455X (gfx1250) — compile-verified
//
#include <hip/hip_runtime.h>

// B=4194304 rows, C=6. Output: single f32 scalar.
// penalty_fn = (t_left ? #rows(useR==0) : 0) + (t_right ? firstL : 0)
//   where firstL = index of first row with useL!=0 (B if none).
// All sigmoid-threshold comparisons done in logit space (monotone).

#define BDIM 256
#define RPT  8
#define NROWS 4194304
#define NB (NROWS / (BDIM * RPT))   // 2048 blocks, exact tile

typedef __attribute__((ext_vector_type(2))) float v2f;
typedef __attribute__((ext_vector_type(8))) float v8f;

struct Accum {
    double   loss;    // 0
    float    cnt;     // 8
    int      firstL;  // 12
    unsigned flags;   // 16  bit0=t_left, bit1=t_right
};

__global__ void ce_init_kernel(Accum* a) {
    a->loss   = 0.0;
    a->cnt    = 0.0f;
    a->firstL = 0x7FFFFFFF;
    a->flags  = 0u;
}

// Wave-wide f32 sum via V_WMMA_F32_16X16X4_F32:
// A = 16x4 matrix holding (x, 0) from each of 32 lanes; B = ones(4x16).
// D[m][n] = rowsum_A(m) for every n, so (sum of my 8 D regs) at lane n plus
// lane n^16 = sum over all 16 rows = sum of ALL A entries = wave sum of x.
// This is independent of the exact A/B lane->element mapping.
__device__ __forceinline__ float wave_sum_wmma(float x) {
    v2f a;    a.x = x;    a.y = 0.0f;
    v2f ones; ones.x = 1.0f; ones.y = 1.0f;
    v8f c = {};
    v8f d = __builtin_amdgcn_wmma_f32_16x16x4_f32(
        /*neg_a=*/false, a, /*neg_b=*/false, ones,
        /*c_mod=*/(short)0, c, /*reuse_a=*/false, /*reuse_b=*/false);
    float s = d[0] + d[1] + d[2] + d[3] + d[4] + d[5] + d[6] + d[7];
    s += __shfl_xor(s, 16, 32);
    return s; // all lanes hold the wave total
}

__device__ __forceinline__ void process_row(
    float p0, float p1, float p2, float p3, float p4, float p5,
    float t0, float t1, float t2, float t3, float t4, float t5,
    int row, float& lossAcc, float& cntAcc, int& minL, bool& tl, bool& tr)
{
    const float T = 0.61903920840622351680f; // log(0.65/0.35)

    // stable log-sum-exp over the 6 logits
    float m = fmaxf(fmaxf(fmaxf(p0, p1), fmaxf(p2, p3)), fmaxf(p4, p5));
    float e = __expf(p0 - m) + __expf(p1 - m) + __expf(p2 - m) +
              __expf(p3 - m) + __expf(p4 - m) + __expf(p5 - m);
    float lse = m + __logf(e);
    float st  = t0 + t1 + t2 + t3 + t4 + t5;
    float dot = p0 * t0 + p1 * t1 + p2 * t2 + p3 * t3 + p4 * t4 + p5 * t5;
    lossAcc += st * lse - dot;   // == -sum(targets * log_softmax)

    // logit-space threshold logic (sigmoid strictly monotone)
    float maxR = fmaxf(fmaxf(p0, p1), p2);
    float maxL = fmaxf(fmaxf(p3, p4), p5);
    bool rset  = maxR > T;
    bool lset  = maxL > T;
    bool rwins = maxR > maxL;
    bool useR  = rset || (!lset && rwins);
    bool useL  = lset || (!rset && !rwins);

    if (!useR) cntAcc += 1.0f;          // left_fn candidate rows
    if (useL)  minL = min(minL, row);   // firstL tracking
    tl = tl || ((t0 + t1 + t2) > 0.0f);
    tr = tr || ((t3 + t4 + t5) > 0.0f);
}

__global__ void __launch_bounds__(BDIM)
ce_main_kernel(const float* __restrict__ preds,
               const float* __restrict__ targets,
               Accum* __restrict__ acc)
{
    const int tid = threadIdx.x;
    const long long row0 = ((long long)blockIdx.x * BDIM + tid) * RPT;

    float lossAcc = 0.0f;
    float cntAcc  = 0.0f;
    int   minL    = 0x7FFFFFFF;
    bool  tl = false, tr = false;

    // 8 consecutive rows per thread = 48 floats = 12 float4 (16B-aligned:
    // byte offset = row0*24 = tid multiple of 192).
    const float4* p4 = reinterpret_cast<const float4*>(preds   + row0 * 6);
    const float4* t4 = reinterpret_cast<const float4*>(targets + row0 * 6);

    #pragma unroll
    for (int it = 0; it < RPT / 2; ++it) {
        float4 pa = p4[it * 3 + 0], pb = p4[it * 3 + 1], pc = p4[it * 3 + 2];
        float4 ta = t4[it * 3 + 0], tb = t4[it * 3 + 1], tc = t4[it * 3 + 2];
        int r = (int)row0 + it * 2;
        process_row(pa.x, pa.y, pa.z, pa.w, pb.x, pb.y,
                    ta.x, ta.y, ta.z, ta.w, tb.x, tb.y,
                    r, lossAcc, cntAcc, minL, tl, tr);
        process_row(pb.z, pb.w, pc.x, pc.y, pc.z, pc.w,
                    tb.z, tb.w, tc.x, tc.y, tc.z, tc.w,
                    r + 1, lossAcc, cntAcc, minL, tl, tr);
    }

    // Wave reductions (all 32 lanes active -> EXEC all-ones for WMMA).
    float waveLoss = wave_sum_wmma(lossAcc);
    float waveCnt  = wave_sum_wmma(cntAcc);
    #pragma unroll
    for (int off = 16; off > 0; off >>= 1)
        minL = min(minL, __shfl_xor(minL, off, 32));
    bool wtl = (__ballot(tl) != 0ull);
    bool wtr = (__ballot(tr) != 0ull);

    __shared__ float sLoss[BDIM / 32];
    __shared__ float sCnt[BDIM / 32];
    __shared__ int   sMin[BDIM / 32];
    __shared__ int   sFlg[BDIM / 32];
    const int wave = tid >> 5;
    const int lane = tid & 31;
    if (lane == 0) {
        sLoss[wave] = waveLoss;
        sCnt[wave]  = waveCnt;
        sMin[wave]  = minL;
        sFlg[wave]  = (wtl ? 1 : 0) | (wtr ? 2 : 0);
    }
    __syncthreads();
    if (tid == 0) {
        float bl = 0.0f, bc = 0.0f;
        int bm = 0x7FFFFFFF, bf = 0;
        #pragma unroll
        for (int w = 0; w < BDIM / 32; ++w) {
            bl += sLoss[w];
            bc += sCnt[w];
            bm  = min(bm, sMin[w]);
            bf |= sFlg[w];
        }
        atomicAdd(&acc->loss, (double)bl);   // global_atomic_add_f64
        atomicAdd(&acc->cnt, bc);            // integer-valued, exact in f32
        atomicMin(&acc->firstL, bm);
        atomicOr(&acc->flags, (unsigned)bf);
    }
}

__global__ void ce_finalize_kernel(const Accum* __restrict__ acc,
                                   float* __restrict__ out)
{
    double lm = acc->loss / (double)NROWS;
    int fl = acc->firstL;
    if (fl > NROWS) fl = NROWS;
    float pen = 0.1f * (((acc->flags & 1u) ? acc->cnt : 0.0f) +
                        ((acc->flags & 2u) ? (float)fl : 0.0f));
    out[0] = (float)lm + pen;
}

extern "C" void kernel_launch(void* const* d_in, const int* in_sizes, int n_in,
                              void* d_out, int out_size, void* d_ws, size_t ws_size,
                              hipStream_t stream)
{
    const float* preds   = (const float*)d_in[0];
    const float* targets = (const float*)d_in[1];
    Accum* acc = (Accum*)d_ws;
    float* out = (float*)d_out;

    hipLaunchKernelGGL(ce_init_kernel, dim3(1), dim3(1), 0, stream, acc);
    hipLaunchKernelGGL(ce_main_kernel, dim3(NB), dim3(BDIM), 0, stream,
                       preds, targets, acc);
    hipLaunchKernelGGL(ce_finalize_kernel, dim3(1), dim3(1), 0, stream, acc, out);
}